// IndexStaged_70128226009354
// MI455X (gfx1250) — compile-verified
//
#include <hip/hip_runtime.h>
#include <hip/hip_bf16.h>

// ---------------------------------------------------------------------------
// MI455X (gfx1250) beam-search ANN.
//  - proj_wmma: exact f32 GEMM q_full = query @ VT^T on V_WMMA_F32_16X16X4_F32
//  - beam_kernel: 1024 threads (32 wave32 = WGP fully occupied) per query.
//    Candidates live in registers (3/thread); top-EF selection = wave32
//    shfl_xor reduce + 32-entry LDS stage (2 barriers/slot); dedup by
//    broadcast-id invalidation. Gather pass is the 23.3 TB/s-bound hot loop.
// ---------------------------------------------------------------------------

typedef __attribute__((ext_vector_type(2))) float v2f;
typedef __attribute__((ext_vector_type(8))) float v8f;

#define DMODEL 128
#define DPRIN  64
#define RNBR   32
#define EFB    64                      // ef_search (hardcoded for sizing)
#define NCAND  (EFB + EFB * RNBR)      // 2112
#define BLK    1024                    // 32 wave32s -> 8 waves/SIMD32
#define NWAVES (BLK / 32)
#define NOWN   3                       // ceil(2112/1024) candidates per thread
#define BIGF   1e30f

// ---------------------------------------------------------------------------
// Kernel 1: q_full[b, j] = sum_k query[b,k] * VT[j,k]   (exact f32 WMMA)
// ---------------------------------------------------------------------------
__global__ __launch_bounds__(32) void proj_wmma(const float* __restrict__ query,
                                                const float* __restrict__ VT,
                                                float* __restrict__ q_full) {
    const int tile = blockIdx.x;          // 64 tiles of 16x16
    const int tm = (tile & 7) * 16;       // batch-row tile
    const int tn = (tile >> 3) * 16;      // output-dim tile
    const int lane = threadIdx.x;
    const int l16 = lane & 15;
    const int kh  = lane >> 4;

    v8f acc = {};
#pragma unroll 4
    for (int k0 = 0; k0 < DMODEL; k0 += 4) {
        v2f a, b;
        const int ka = k0 + 2 * kh;
        a.x = query[(size_t)(tm + l16) * DMODEL + ka + 0];
        a.y = query[(size_t)(tm + l16) * DMODEL + ka + 1];
        b.x = VT[(size_t)(tn + l16) * DMODEL + ka + 0];   // B[k][n] = VT[n][k]
        b.y = VT[(size_t)(tn + l16) * DMODEL + ka + 1];
        acc = __builtin_amdgcn_wmma_f32_16x16x4_f32(
            false, a, false, b, (short)0, acc, false, false);
    }
#pragma unroll
    for (int r = 0; r < 8; ++r)
        q_full[(size_t)(tm + r + 8 * kh) * DMODEL + tn + l16] = acc[r];
}

// ---------------------------------------------------------------------------
// Principal-subspace distance: ||s_p||^2 - 2 q_p.s_p + ||q_p||^2
// ---------------------------------------------------------------------------
__device__ __forceinline__ float dist_p(int id, const float* __restrict__ storage,
                                        const float* qf, float qn, int n) {
    if ((unsigned)id >= (unsigned)n) return BIGF;
    const float4* __restrict__ row = (const float4*)(storage + (size_t)id * DMODEL);
    const float4* qv = (const float4*)qf;
    float dot = 0.f, ss = 0.f;
#pragma unroll
    for (int i = 0; i < DPRIN / 4; ++i) {
        float4 v = row[i];
        float4 q = qv[i];
        dot += v.x * q.x + v.y * q.y + v.z * q.z + v.w * q.w;
        ss  += v.x * v.x + v.y * v.y + v.z * v.z + v.w * v.w;
    }
    return ss - 2.f * dot + qn;
}

// ---------------------------------------------------------------------------
// Register-resident dedup top-M.  Each thread owns NO candidates in regs.
// Per slot: 3-reg local min -> wave32 shfl_xor (dist,id) min-reduce ->
// 32 wave partials in LDS -> wave0 shuffle-reduce -> broadcast winner ->
// all threads invalidate matching ids in their registers.
// Duplicated ids always carry identical distances, so this reproduces the
// reference's sort-by-id dedup + top_k.
// ---------------------------------------------------------------------------
template <int NO>
__device__ void select_topm_reg(int out_m, int* rid, float* rdist,
                                int* beamI, float* beamD,
                                float* wredD, int* wredI,
                                int* selId, int tid, int n) {
    const int lane = tid & 31;
    const int wave = tid >> 5;
    for (int slot = 0; slot < out_m; ++slot) {
        float best = BIGF;
        int   bid  = n;
#pragma unroll
        for (int s = 0; s < NO; ++s)
            if (rdist[s] < best) { best = rdist[s]; bid = rid[s]; }
#pragma unroll
        for (int m = 16; m > 0; m >>= 1) {
            float od = __shfl_xor(best, m, 32);
            int   oi = __shfl_xor(bid,  m, 32);
            if (od < best) { best = od; bid = oi; }
        }
        if (lane == 0) { wredD[wave] = best; wredI[wave] = bid; }
        __syncthreads();
        if (wave == 0) {
            float wd = (lane < NWAVES) ? wredD[lane] : BIGF;
            int   wi = (lane < NWAVES) ? wredI[lane] : n;
#pragma unroll
            for (int m = 16; m > 0; m >>= 1) {
                float od = __shfl_xor(wd, m, 32);
                int   oi = __shfl_xor(wi, m, 32);
                if (od < wd) { wd = od; wi = oi; }
            }
            if (lane == 0) {
                if (wd >= BIGF) { wi = n; wd = BIGF; }   // padding slot
                *selId = wi;
                beamI[slot] = wi;
                beamD[slot] = wd;
            }
        }
        __syncthreads();
        const int sid = *selId;
        if (sid < n) {
#pragma unroll
            for (int s = 0; s < NO; ++s)
                if (rid[s] == sid) rdist[s] = BIGF;
        }
    }
}

// ---------------------------------------------------------------------------
// Kernel 2: full beam search, one 1024-thread workgroup per query.
// ---------------------------------------------------------------------------
__global__ __launch_bounds__(BLK) void beam_kernel(
        const float* __restrict__ storage,
        const int*   __restrict__ neighbors,
        const int*   __restrict__ entry_ids,
        const float* __restrict__ q_full,
        int*   __restrict__ out_ids,
        float* __restrict__ out_d,
        int n, int e_count, int k, int n_iters) {
    __shared__ float qf[DMODEL];
    __shared__ int   beamI[EFB];
    __shared__ float beamD[EFB];
    __shared__ float wredD[NWAVES];
    __shared__ int   wredI[NWAVES];
    __shared__ int   selId;

    const int b   = blockIdx.x;
    const int tid = threadIdx.x;

    for (int i = tid; i < DMODEL; i += BLK)
        qf[i] = q_full[(size_t)b * DMODEL + i];
    __syncthreads();

    float qn = 0.f;
#pragma unroll
    for (int i = 0; i < DPRIN; ++i) qn += qf[i] * qf[i];

    int   rid[NOWN];
    float rdist[NOWN];

    // ---- init beam from entry points (thread tid owns candidate tid) ----
#pragma unroll
    for (int s = 0; s < NOWN; ++s) { rid[s] = n; rdist[s] = BIGF; }
    if (tid < e_count) {
        rid[0]   = entry_ids[tid];
        rdist[0] = dist_p(rid[0], storage, qf, qn, n);
    }
    __syncthreads();
    select_topm_reg<NOWN>(EFB, rid, rdist, beamI, beamD, wredD, wredI,
                          &selId, tid, n);

    // ---- beam iterations ------------------------------------------------
    for (int t = 0; t < n_iters; ++t) {
        // capture candidate ids into registers; prefetch their rows
#pragma unroll
        for (int s = 0; s < NOWN; ++s) {
            const int j = tid + s * BLK;
            int id = n;
            float d = BIGF;
            if (j < EFB) {                      // carried beam entry
                id = beamI[j];
                d  = beamD[j];
            } else if (j < NCAND) {             // neighbor expansion
                const int node = beamI[(j - EFB) >> 5];
                id = (node >= n) ? n
                                 : neighbors[(size_t)node * RNBR + ((j - EFB) & 31)];
                if ((unsigned)id < (unsigned)n)
                    __builtin_prefetch(storage + (size_t)id * DMODEL, 0, 1);
            }
            rid[s]   = id;
            rdist[s] = d;
        }
        // distance pass (the HBM-gather hot loop)
#pragma unroll
        for (int s = 0; s < NOWN; ++s) {
            const int j = tid + s * BLK;
            if (j >= EFB && j < NCAND)
                rdist[s] = dist_p(rid[s], storage, qf, qn, n);
        }
        __syncthreads();   // all captures done before beam is overwritten
        select_topm_reg<NOWN>(EFB, rid, rdist, beamI, beamD, wredD, wredI,
                              &selId, tid, n);
    }

    // ---- final exact 128-dim re-rank of the beam ------------------------
    {
        int   fid = n;
        float fd  = BIGF;
        if (tid < EFB) {
            fid = beamI[tid];
            if ((unsigned)fid < (unsigned)n) {
                const float4* __restrict__ row =
                    (const float4*)(storage + (size_t)fid * DMODEL);
                const float4* qv = (const float4*)qf;
                fd = 0.f;
#pragma unroll
                for (int i = 0; i < DMODEL / 4; ++i) {
                    float4 v = row[i];
                    float4 q = qv[i];
                    float dx = v.x - q.x, dy = v.y - q.y;
                    float dz = v.z - q.z, dw = v.w - q.w;
                    fd += dx * dx + dy * dy + dz * dz + dw * dw;
                }
            }
        }
        __syncthreads();   // captures complete before selection writes beam
        int   rid1[1]   = { fid };
        float rdist1[1] = { fd };
        select_topm_reg<1>(k, rid1, rdist1, beamI, beamD, wredD, wredI,
                           &selId, tid, n);
    }
    if (tid < k) {
        out_ids[(size_t)b * k + tid] = beamI[tid];
        out_d[(size_t)b * k + tid]   = beamD[tid];
    }
}

// ---------------------------------------------------------------------------
// Host launcher.  Inputs: query, VT, storage, neighbors, entry_ids, k, ef.
// d_out: [B*K] int32 ids then [B*K] f32 dists (tuple concatenated flat).
// ---------------------------------------------------------------------------
extern "C" void kernel_launch(void* const* d_in, const int* in_sizes, int n_in,
                              void* d_out, int out_size, void* d_ws, size_t ws_size,
                              hipStream_t stream) {
    const float* query     = (const float*)d_in[0];
    const float* VT        = (const float*)d_in[1];
    const float* storage   = (const float*)d_in[2];
    const int*   neighbors = (const int*)d_in[3];
    const int*   entry_ids = (const int*)d_in[4];
    const int B = in_sizes[0] / DMODEL;          // 128
    const int n = in_sizes[2] / DMODEL;          // 1,000,000
    const int e_count = in_sizes[4];             // 32
    const int k = out_size / (2 * B);            // 10
    const int n_iters = 8;

    float* q_full  = (float*)d_ws;               // B * 128 floats
    int*   out_ids = (int*)d_out;
    float* out_d   = (float*)d_out + (size_t)B * k;

    proj_wmma<<<(B / 16) * (DMODEL / 16), 32, 0, stream>>>(query, VT, q_full);
    beam_kernel<<<B, BLK, 0, stream>>>(storage, neighbors, entry_ids, q_full,
                                       out_ids, out_d, n, e_count, k, n_iters);
}